// MMSolver_1640677507629
// MI455X (gfx1250) — compile-verified
//
#include <hip/hip_runtime.h>
#include <stdint.h>

namespace {

constexpr int kNX = 128, kNY = 128, kNCell = kNX * kNY;
constexpr int kT = 256, kNSrc = 2, kNProbe = 4, kRelax = 100;
constexpr int kThreads = 1024;              // 32 wave32 waves -> 8 per SIMD32, one WGP
constexpr int kCPT = kNCell / kThreads;     // 16 cells per thread

constexpr float kMu0        = 1.25663706143591729e-6f;   // 4e-7 * pi
constexpr float kAex        = 1.3e-11f;
constexpr float kDx         = 5e-9f;
constexpr float kH          = 175950000000.0f * 1e-13f;  // GAMMA*DT = 0.017595
constexpr float kAlphaRun   = 0.02f;
constexpr float kAlphaRelax = 0.5f;
constexpr float kInvRun     = 1.0f / (1.0f + kAlphaRun * kAlphaRun);
constexpr float kInvRelax   = 1.0f / (1.0f + kAlphaRelax * kAlphaRelax);  // 0.8 exact
constexpr float kCSot       = 0.001f;

// Compiler scheduling fence: caps how many cells' loads get hoisted at once.
#define SCHED_FENCE() asm volatile("" ::: "memory")

// LDS plan (exactly 320 KiB = full CDNA5 WGP LDS):
//   s_m  : kNCell float4  {mx, my, mz, mu0*Msat}   256 KiB  (w written once, preserved)
//   s_ce : kNCell float   2*Aex/(Msat*dx^2)         64 KiB  (DMA'd as Msat, converted in place)
__global__ __launch_bounds__(kThreads, 1)
void mmsolver_kernel(const float* __restrict__ signal,
                     const float* __restrict__ bext,
                     const float* __restrict__ msat,
                     const int*   __restrict__ src_idx,
                     const int*   __restrict__ probe_idx,
                     float*       __restrict__ out)
{
  extern __shared__ float smem[];
  float* s_m  = smem;              // [kNCell][4]
  float* s_ce = smem + 4 * kNCell; // [kNCell]

  const int tid = threadIdx.x;

  // ---- gfx1250 async global->LDS DMA of Msat into the cE table region ----
#pragma unroll
  for (int i = 0; i < kCPT; ++i) {
    const int idx = tid + kThreads * i;
    uint32_t lds_off = (uint32_t)(uintptr_t)(&s_ce[idx]);
    uint64_t gaddr   = (uint64_t)(uintptr_t)(&msat[idx]);
    asm volatile("global_load_async_to_lds_b32 %0, %1, off"
                 :: "v"(lds_off), "v"(gaddr) : "memory");
  }
  asm volatile("s_wait_asynccnt 0x0" ::: "memory");
  __syncthreads();

  // Uniform scalars (scalar loads).
  const int src0 = src_idx[0] * kNY + src_idx[1];
  const int src1 = src_idx[2] * kNY + src_idx[3];

  // One probe per low thread: everything it needs in two registers.
  int   myprobe = 0;
  float pms = 0.0f, pm0 = 0.0f;
  if (tid < kNProbe) {
    myprobe = probe_idx[2 * tid] * kNY + probe_idx[2 * tid + 1];
    pms     = msat[myprobe];
  }

  // Per-cell persistent register state: m at start of step only.
  float mx[kCPT], my[kCPT], mz[kCPT];
#pragma unroll
  for (int i = 0; i < kCPT; ++i) {
    const int idx = tid + kThreads * i;
    const float ms = s_ce[idx];                       // DMA'd Msat (own cell, own wait)
    s_ce[idx] = 2.0f * kAex / (ms * kDx * kDx);       // in-place -> cE table
    *reinterpret_cast<float4*>(&s_m[4 * idx]) =
        make_float4(0.0f, 1.0f, 0.0f, kMu0 * ms);     // m=(0,1,0), w = demag coeff
    mx[i] = 0.0f; my[i] = 1.0f; mz[i] = 0.0f;
  }
  __syncthreads();

  // LLG torque at the staged field in LDS (one ds_load_b128 per center/neighbor).
  auto torque = [&](int i, float sig0, float sig1, bool inject,
                    float alpha, float inv1a,
                    float& kx, float& ky, float& kz) {
    const int idx = tid + kThreads * i;
    const int x = idx >> 7, y = idx & (kNY - 1);
    const int iN = (x > 0)       ? idx - kNY : idx;   // edge-clamped 5-point stencil
    const int iS = (x < kNX - 1) ? idx + kNY : idx;
    const int iW = (y > 0)       ? idx - 1   : idx;
    const int iE = (y < kNY - 1) ? idx + 1   : idx;
    const float4 C = *reinterpret_cast<const float4*>(&s_m[4 * idx]);
    const float4 N = *reinterpret_cast<const float4*>(&s_m[4 * iN]);
    const float4 S = *reinterpret_cast<const float4*>(&s_m[4 * iS]);
    const float4 W = *reinterpret_cast<const float4*>(&s_m[4 * iW]);
    const float4 E = *reinterpret_cast<const float4*>(&s_m[4 * iE]);
    const float cEx = s_ce[idx];
    const float lx = N.x + S.x + W.x + E.x - 4.0f * C.x;
    const float ly = N.y + S.y + W.y + E.y - 4.0f * C.y;
    const float lz = N.z + S.z + W.z + E.z - 4.0f * C.z;
    const float bx = bext[0 * kNCell + idx] + cEx * lx;
    const float by = bext[1 * kNCell + idx] + cEx * ly;
    float bz0 = bext[2 * kNCell + idx];
    if (inject) {                       // B_ext.at[0,2,src].set(sig)
      if (idx == src0) bz0 = sig0;
      if (idx == src1) bz0 = sig1;
    }
    const float bz = bz0 + cEx * lz - C.w * C.z;      // + exchange - demag(z)
    // t1 = m x B
    const float t1x = C.y * bz - C.z * by;
    const float t1y = C.z * bx - C.x * bz;
    const float t1z = C.x * by - C.y * bx;
    // t2 = m x (m x B)
    const float t2x = C.y * t1z - C.z * t1y;
    const float t2y = C.z * t1x - C.x * t1z;
    const float t2z = C.x * t1y - C.y * t1x;
    // sot = C_SOT * m x (m x yhat) = C_SOT * (mx*my, -(mx^2+mz^2), my*mz)
    kx = -inv1a * (t1x + alpha * t2x) + kCSot * (C.x * C.y);
    ky = -inv1a * (t1y + alpha * t2y) - kCSot * (C.x * C.x + C.z * C.z);
    kz = -inv1a * (t1z + alpha * t2z) + kCSot * (C.y * C.z);
  };

  // ---- unified time loop: 100 relax + 256 driven steps ----
  for (int it = 0; it < kRelax + kT; ++it) {
    const bool  driven = (it >= kRelax);
    const float alpha  = driven ? kAlphaRun : kAlphaRelax;
    const float inv1a  = driven ? kInvRun   : kInvRelax;
    float sig0 = 0.0f, sig1 = 0.0f;
    if (driven) {                         // uniform scalar loads, L2 resident
      sig0 = signal[2 * (it - kRelax) + 0];
      sig1 = signal[2 * (it - kRelax) + 1];
    }

    float ax[kCPT], ay[kCPT], az[kCPT];   // k1 + 2k2 + 2k3 + k4
    float tx[kCPT], ty[kCPT], tz[kCPT];   // current-stage k

    // k1
#pragma unroll
    for (int i = 0; i < kCPT; ++i) {
      torque(i, sig0, sig1, driven, alpha, inv1a, tx[i], ty[i], tz[i]);
      ax[i] = tx[i]; ay[i] = ty[i]; az[i] = tz[i];
      if ((i & 1) == 1) SCHED_FENCE();
    }
    __syncthreads();
#pragma unroll
    for (int i = 0; i < kCPT; ++i) {
      const int idx = tid + kThreads * i;
      s_m[4 * idx + 0] = mx[i] + 0.5f * kH * tx[i];   // .w (demag coeff) preserved
      s_m[4 * idx + 1] = my[i] + 0.5f * kH * ty[i];
      s_m[4 * idx + 2] = mz[i] + 0.5f * kH * tz[i];
      if ((i & 3) == 3) SCHED_FENCE();
    }
    __syncthreads();
    // k2
#pragma unroll
    for (int i = 0; i < kCPT; ++i) {
      torque(i, sig0, sig1, driven, alpha, inv1a, tx[i], ty[i], tz[i]);
      ax[i] += 2.0f * tx[i]; ay[i] += 2.0f * ty[i]; az[i] += 2.0f * tz[i];
      if ((i & 1) == 1) SCHED_FENCE();
    }
    __syncthreads();
#pragma unroll
    for (int i = 0; i < kCPT; ++i) {
      const int idx = tid + kThreads * i;
      s_m[4 * idx + 0] = mx[i] + 0.5f * kH * tx[i];
      s_m[4 * idx + 1] = my[i] + 0.5f * kH * ty[i];
      s_m[4 * idx + 2] = mz[i] + 0.5f * kH * tz[i];
      if ((i & 3) == 3) SCHED_FENCE();
    }
    __syncthreads();
    // k3
#pragma unroll
    for (int i = 0; i < kCPT; ++i) {
      torque(i, sig0, sig1, driven, alpha, inv1a, tx[i], ty[i], tz[i]);
      ax[i] += 2.0f * tx[i]; ay[i] += 2.0f * ty[i]; az[i] += 2.0f * tz[i];
      if ((i & 1) == 1) SCHED_FENCE();
    }
    __syncthreads();
#pragma unroll
    for (int i = 0; i < kCPT; ++i) {
      const int idx = tid + kThreads * i;
      s_m[4 * idx + 0] = mx[i] + kH * tx[i];
      s_m[4 * idx + 1] = my[i] + kH * ty[i];
      s_m[4 * idx + 2] = mz[i] + kH * tz[i];
      if ((i & 3) == 3) SCHED_FENCE();
    }
    __syncthreads();
    // k4
#pragma unroll
    for (int i = 0; i < kCPT; ++i) {
      torque(i, sig0, sig1, driven, alpha, inv1a, tx[i], ty[i], tz[i]);
      ax[i] += tx[i]; ay[i] += ty[i]; az[i] += tz[i];
      if ((i & 1) == 1) SCHED_FENCE();
    }
    __syncthreads();
#pragma unroll
    for (int i = 0; i < kCPT; ++i) {
      const int idx = tid + kThreads * i;
      mx[i] += (kH / 6.0f) * ax[i];
      my[i] += (kH / 6.0f) * ay[i];
      mz[i] += (kH / 6.0f) * az[i];
      s_m[4 * idx + 0] = mx[i];
      s_m[4 * idx + 1] = my[i];
      s_m[4 * idx + 2] = mz[i];
      if ((i & 3) == 3) SCHED_FENCE();
    }
    __syncthreads();

    // Probe work: threads 0..3 own one probe each; s_m is stable in this
    // window (next write to s_m is separated by the k1 compute + barrier).
    if (tid < kNProbe) {
      if (it == kRelax - 1) pm0 = s_m[4 * myprobe + 2];      // m0_z snapshot
      if (driven)
        out[(it - kRelax) * kNProbe + tid] = (s_m[4 * myprobe + 2] - pm0) * pms;
    }
  }
}

} // namespace

extern "C" void kernel_launch(void* const* d_in, const int* in_sizes, int n_in,
                              void* d_out, int out_size, void* d_ws, size_t ws_size,
                              hipStream_t stream) {
  (void)in_sizes; (void)n_in; (void)out_size; (void)d_ws; (void)ws_size;
  const float* signal    = (const float*)d_in[0];
  const float* bext      = (const float*)d_in[1];
  const float* msat      = (const float*)d_in[2];
  const int*   src_idx   = (const int*)d_in[3];
  const int*   probe_idx = (const int*)d_in[4];
  float* out = (float*)d_out;

  constexpr size_t kSmemBytes = (size_t)(4 * kNCell + kNCell) * sizeof(float); // 327,680 B = 320 KiB
  (void)hipFuncSetAttribute(reinterpret_cast<const void*>(mmsolver_kernel),
                            hipFuncAttributeMaxDynamicSharedMemorySize,
                            (int)kSmemBytes);
  mmsolver_kernel<<<dim3(1), dim3(kThreads), kSmemBytes, stream>>>(
      signal, bext, msat, src_idx, probe_idx, out);
}